// GATModel_2078764171902
// MI455X (gfx1250) — compile-verified
//
#include <hip/hip_runtime.h>

#define NNODES 65536
#define NEDGES 1048576
#define INC    128
#define HIDC   32
#define NHEADS 4
#define OUTC   64
#define NEG_SLOPE 0.2f
#define EPSV   1e-16f

typedef float v2f __attribute__((ext_vector_type(2)));
typedef float v8f __attribute__((ext_vector_type(8)));

// ---------------------------------------------------------------------------
// f32 WMMA GEMM: C[M, NT*16] = A[M,K] @ B[K, NT*16], row-major.
// One wave per 16-row block; the wave keeps the A fragment in registers and
// sweeps all NT column tiles (A traffic cut by NT, NT wmma ops per K-step).
// V_WMMA_F32_16X16X4_F32 layouts (CDNA5, wave32):
//  A frag (16x4): lanes0-15 M=0..15 v0=K0/v1=K1; lanes16-31 v0=K2/v1=K3
//  B frag (4x16): lanes0-15 N=0..15 v0=K0/v1=K1; lanes16-31 v0=K2/v1=K3
//  D (16x16, 8 VGPR): vgpr r -> row r (lanes0-15) / r+8 (lanes16-31), col=lane&15
// ---------------------------------------------------------------------------
template <int NT>
__global__ void __launch_bounds__(32)
wmma_gemm_f32(const float* __restrict__ A, const float* __restrict__ B,
              float* __restrict__ C, int K) {
    const int Ncols = NT * 16;
    const int lane = threadIdx.x & 31;
    const int half = lane >> 4;     // 0 or 1
    const int l15  = lane & 15;
    const int row  = (blockIdx.x << 4) + l15;   // A row for this lane
    v8f acc[NT];
#pragma unroll
    for (int t = 0; t < NT; ++t) acc[t] = (v8f){};

    for (int k0 = 0; k0 < K; k0 += 4) {
        const int ka = k0 + 2 * half;
        v2f a;
        a.x = A[(size_t)row * K + ka];
        a.y = A[(size_t)row * K + ka + 1];
#pragma unroll
        for (int t = 0; t < NT; ++t) {
            v2f b;
            b.x = B[(size_t)ka * Ncols + t * 16 + l15];
            b.y = B[(size_t)(ka + 1) * Ncols + t * 16 + l15];
            acc[t] = __builtin_amdgcn_wmma_f32_16x16x4_f32(
                false, a, false, b, (short)0, acc[t], false, false);
        }
    }
    const int orow = (blockIdx.x << 4) + (half << 3);
#pragma unroll
    for (int t = 0; t < NT; ++t)
#pragma unroll
        for (int r = 0; r < 8; ++r)
            C[(size_t)(orow + r) * Ncols + t * 16 + l15] = acc[t][r];
}

// ---------------------------------------------------------------------------
// Per-(node,head) attention scores: a_src[n,h] = h[n,h,:]·att_src[h,:], etc.
// ---------------------------------------------------------------------------
__global__ void att_scores(const float* __restrict__ hfeat,
                           const float* __restrict__ att_src,
                           const float* __restrict__ att_dst,
                           float* __restrict__ a_src, float* __restrict__ a_dst,
                           int Nn, int H, int Cc) {
    int idx = blockIdx.x * blockDim.x + threadIdx.x;   // n*H + h
    if (idx >= Nn * H) return;
    int n = idx / H, hh = idx - n * H;
    const float* hp = hfeat + (size_t)n * H * Cc + (size_t)hh * Cc;
    const float* as = att_src + (size_t)hh * Cc;
    const float* ad = att_dst + (size_t)hh * Cc;
    float s = 0.f, d = 0.f;
    for (int c = 0; c < Cc; ++c) { float v = hp[c]; s += v * as[c]; d += v * ad[c]; }
    a_src[idx] = s;
    a_dst[idx] = d;
}

// float <-> order-preserving uint key (so unsigned atomicMax == float max)
__device__ __forceinline__ unsigned f2key(float f) {
    unsigned b = __float_as_uint(f);
    return (b & 0x80000000u) ? ~b : (b | 0x80000000u);
}
__device__ __forceinline__ float key2f(unsigned k) {
    unsigned b = (k & 0x80000000u) ? (k & 0x7FFFFFFFu) : ~k;
    return __uint_as_float(b);
}
__device__ __forceinline__ float lrelu(float x) {
    return x > 0.f ? x : x * NEG_SLOPE;
}

// Pass A: segment max of leaky-relu'd logits into amax (uint-key atomicMax)
__global__ void edge_max(const long long* __restrict__ ei,
                         const float* __restrict__ a_src,
                         const float* __restrict__ a_dst,
                         unsigned* __restrict__ amax, int Ecnt, int H) {
    int e = blockIdx.x * blockDim.x + threadIdx.x;
    if (e >= Ecnt) return;
    int s = (int)ei[e];
    int d = (int)ei[(size_t)Ecnt + e];
    for (int h = 0; h < H; ++h) {
        float al = lrelu(a_src[s * H + h] + a_dst[d * H + h]);
        atomicMax(&amax[d * H + h], f2key(al));
    }
}

// Pass B: denom[dst,h] += exp(alpha - amax[dst,h])
__global__ void edge_expsum(const long long* __restrict__ ei,
                            const float* __restrict__ a_src,
                            const float* __restrict__ a_dst,
                            const unsigned* __restrict__ amax,
                            float* __restrict__ denom, int Ecnt, int H) {
    int e = blockIdx.x * blockDim.x + threadIdx.x;
    if (e >= Ecnt) return;
    int s = (int)ei[e];
    int d = (int)ei[(size_t)Ecnt + e];
    for (int h = 0; h < H; ++h) {
        float al = lrelu(a_src[s * H + h] + a_dst[d * H + h]);
        float m  = key2f(amax[d * H + h]);
        atomicAdd(&denom[d * H + h], __expf(al - m));
    }
}

// ---------------------------------------------------------------------------
// Pass C: agg[dst, :] += coef(e,head) * hfeat[src, :].
// 32 lanes per edge; each lane owns VEC contiguous channels so the feature
// gather is a single b128/b64 load and the softmax coefficient is computed
// once per lane (lane*VEC never crosses a head boundary for these shapes).
// ---------------------------------------------------------------------------
template <int VEC, int CC, int HH>
__global__ void edge_aggregate(const long long* __restrict__ ei,
                               const float* __restrict__ a_src,
                               const float* __restrict__ a_dst,
                               const unsigned* __restrict__ amax,
                               const float* __restrict__ denom,
                               const float* __restrict__ hfeat,
                               float* __restrict__ agg, int Ecnt) {
    int gid  = blockIdx.x * blockDim.x + threadIdx.x;
    int e    = gid >> 5;
    int lane = gid & 31;
    if (e >= Ecnt) return;
    int s = (int)ei[e];
    int d = (int)ei[(size_t)Ecnt + e];
    const int F  = 32 * VEC;          // = HH * CC
    const int f0 = lane * VEC;
    const int hh = f0 / CC;           // head owning all VEC channels of this lane

    float al   = lrelu(a_src[s * HH + hh] + a_dst[d * HH + hh]);
    float m    = key2f(amax[d * HH + hh]);
    float coef = __expf(al - m) / (denom[d * HH + hh] + EPSV);

    const float* hp = hfeat + (size_t)s * F + f0;
    float*       ap = agg   + (size_t)d * F + f0;

    float vals[VEC];
    if constexpr (VEC == 4) {
        float4 v = *(const float4*)hp;
        vals[0] = v.x; vals[1] = v.y; vals[2] = v.z; vals[3] = v.w;
    } else {
        float2 v = *(const float2*)hp;
        vals[0] = v.x; vals[1] = v.y;
    }
#pragma unroll
    for (int j = 0; j < VEC; ++j)
        atomicAdd(ap + j, coef * vals[j]);
}

// out[i] = sigmoid(agg[i] + bias[i % F])
__global__ void bias_sigmoid(const float* __restrict__ agg,
                             const float* __restrict__ bias,
                             float* __restrict__ out, int total, int F) {
    int i = blockIdx.x * blockDim.x + threadIdx.x;
    if (i >= total) return;
    float v = agg[i] + bias[i % F];
    out[i] = 1.f / (1.f + __expf(-v));
}

// ---------------------------------------------------------------------------
extern "C" void kernel_launch(void* const* d_in, const int* in_sizes, int n_in,
                              void* d_out, int out_size, void* d_ws, size_t ws_size,
                              hipStream_t stream) {
    const float*     x   = (const float*)d_in[0];
    const long long* ei  = (const long long*)d_in[1];   // int64 [2, E]
    const float*     W1  = (const float*)d_in[2];
    const float*     as1 = (const float*)d_in[3];
    const float*     ad1 = (const float*)d_in[4];
    const float*     b1  = (const float*)d_in[5];
    const float*     W2  = (const float*)d_in[6];
    const float*     as2 = (const float*)d_in[7];
    const float*     ad2 = (const float*)d_in[8];
    const float*     b2  = (const float*)d_in[9];
    float* out = (float*)d_out;

    const size_t N = NNODES;
    // Workspace layout (floats): ~84 MB total
    float*    h     = (float*)d_ws;               // N*128 (h1, later reused for h2)
    float*    agg1  = h + N * 128;                // N*128 (agg1 -> x2 in place)
    float*    a_src = agg1 + N * 128;             // N*4
    float*    a_dst = a_src + N * 4;              // N*4
    unsigned* amax  = (unsigned*)(a_dst + N * 4); // N*4
    float*    denom = (float*)amax + N * 4;       // N*4
    float*    agg2  = denom + N * 4;              // N*64

    const int TB = 256;

    // ==================== Layer 1 (H=4, C=32, F=128) ====================
    hipMemsetAsync(agg1, 0, N * 128 * sizeof(float), stream);
    hipMemsetAsync(amax, 0, N * 4 * sizeof(unsigned) * 2, stream); // amax+denom

    wmma_gemm_f32<8><<<dim3(N / 16), 32, 0, stream>>>(x, W1, h, INC);
    att_scores<<<(N * NHEADS + TB - 1) / TB, TB, 0, stream>>>(h, as1, ad1, a_src, a_dst,
                                                              (int)N, NHEADS, HIDC);
    edge_max<<<(NEDGES + TB - 1) / TB, TB, 0, stream>>>(ei, a_src, a_dst, amax, NEDGES, NHEADS);
    edge_expsum<<<(NEDGES + TB - 1) / TB, TB, 0, stream>>>(ei, a_src, a_dst, amax, denom,
                                                           NEDGES, NHEADS);
    edge_aggregate<4, HIDC, NHEADS><<<(NEDGES * 32) / TB, TB, 0, stream>>>(
        ei, a_src, a_dst, amax, denom, h, agg1, NEDGES);
    bias_sigmoid<<<(N * 128 + TB - 1) / TB, TB, 0, stream>>>(agg1, b1, agg1,
                                                             (int)(N * 128), 128);

    // ==================== Layer 2 (H=1, C=64, F=64) ====================
    hipMemsetAsync(agg2, 0, N * 64 * sizeof(float), stream);
    hipMemsetAsync(amax, 0, N * 4 * sizeof(unsigned) * 2, stream);

    wmma_gemm_f32<4><<<dim3(N / 16), 32, 0, stream>>>(agg1, W2, h, NHEADS * HIDC);
    att_scores<<<(N + TB - 1) / TB, TB, 0, stream>>>(h, as2, ad2, a_src, a_dst, (int)N, 1, OUTC);
    edge_max<<<(NEDGES + TB - 1) / TB, TB, 0, stream>>>(ei, a_src, a_dst, amax, NEDGES, 1);
    edge_expsum<<<(NEDGES + TB - 1) / TB, TB, 0, stream>>>(ei, a_src, a_dst, amax, denom,
                                                           NEDGES, 1);
    edge_aggregate<2, OUTC, 1><<<(NEDGES * 32) / TB, TB, 0, stream>>>(
        ei, a_src, a_dst, amax, denom, h, agg2, NEDGES);
    bias_sigmoid<<<(N * 64 + TB - 1) / TB, TB, 0, stream>>>(agg2, b2, out,
                                                            (int)(N * 64), 64);
    (void)in_sizes; (void)n_in; (void)out_size; (void)ws_size;
}